// PIQE_61409442398707
// MI455X (gfx1250) — compile-verified
//
#include <hip/hip_runtime.h>
#include <math.h>

typedef __attribute__((ext_vector_type(2))) float v2f;
typedef __attribute__((ext_vector_type(8))) float v8f;

#define IMG_H 2048
#define IMG_W 2048
#define NB    128            // blocks per side
#define NPIX  (IMG_H * IMG_W)
#define NBLK  (NB * NB)      // 16384 blocks per batch
#define NBATCH 8

// ---------------------------------------------------------------- utilities
__device__ __forceinline__ float wave_sum32(float v) {
#pragma unroll
  for (int m = 16; m > 0; m >>= 1) v += __shfl_xor(v, m, 32);
  return v;
}

// symmetric gaussian tap g[d], d in [0,6]; 0 outside. g symmetric: g[d]=g[6-d]
__device__ __forceinline__ float gsel(int d, float g0, float g1, float g2, float g3) {
  float r = 0.0f;
  r = (d == 0 || d == 6) ? g0 : r;
  r = (d == 1 || d == 5) ? g1 : r;
  r = (d == 2 || d == 4) ? g2 : r;
  r = (d == 3)           ? g3 : r;
  return r;
}

// ---------------------------------------------------------------- kernels
__global__ void piqe_init_kernel(unsigned* maxBits) {
  if (threadIdx.x < NBATCH) maxBits[threadIdx.x] = 0u;
}

__global__ void piqe_max_kernel(const float* __restrict__ x, unsigned* maxBits) {
  const int b = blockIdx.y;
  const float* xb = x + (size_t)b * NPIX;
  unsigned m = 0u;
  for (int i = blockIdx.x * blockDim.x + threadIdx.x; i < NPIX;
       i += gridDim.x * blockDim.x) {
    // inputs are uniform[0,1): non-negative, so uint bit order == float order
    m = max(m, __float_as_uint(xb[i]));
  }
  __shared__ unsigned sm[256];
  sm[threadIdx.x] = m;
  __syncthreads();
  for (int s = 128; s > 0; s >>= 1) {
    if (threadIdx.x < s) sm[threadIdx.x] = max(sm[threadIdx.x], sm[threadIdx.x + s]);
    __syncthreads();
  }
  if (threadIdx.x == 0) atomicMax(&maxBits[b], sm[0]);
}

// One wave32 per 16x16 image block.
__global__ __launch_bounds__(32)
void piqe_block_kernel(const float* __restrict__ x,
                       const unsigned* __restrict__ maxBits,
                       float* __restrict__ contribBuf,
                       float* __restrict__ activeBuf) {
  __shared__ float imgT[22 * 24];   // 22x22 halo tile, stride 24
  __shared__ float Hmu[24 * 16];    // horizontal-pass mu   (rows 22,23 zero)
  __shared__ float Hmu2[24 * 16];   // horizontal-pass mu2
  __shared__ float nbuf[16 * 17];   // normalized block, stride 17

  const int lane = threadIdx.x;
  const int bx = blockIdx.x, by = blockIdx.y, bz = blockIdx.z;
  const float* xb = x + (size_t)bz * NPIX;

  // gaussian 1D taps, sigma = 7/6  ->  w(d) = exp(-d^2 * 18/49)
  const float w1 = expf(-18.0f / 49.0f);
  const float w2 = expf(-72.0f / 49.0f);
  const float w3 = expf(-162.0f / 49.0f);
  const float inv = 1.0f / (1.0f + 2.0f * (w1 + w2 + w3));
  const float g0 = w3 * inv, g1 = w2 * inv, g2 = w1 * inv, g3 = inv;

  const float scale = 255.0f / __uint_as_float(maxBits[bz]);

  // ---- load 22x22 halo with edge clamp, quantize (round half-even) --------
  const int baseR = by * 16 - 3, baseC = bx * 16 - 3;
#pragma unroll
  for (int it = 0; it < 16; ++it) {
    int idx = lane + 32 * it;
    if (idx < 484) {
      int r = idx / 22, c = idx - r * 22;
      int gr = min(max(baseR + r, 0), IMG_H - 1);
      int gc = min(max(baseC + c, 0), IMG_W - 1);
      imgT[r * 24 + c] = rintf(scale * xb[(size_t)gr * IMG_W + gc]);
    }
  }
  __syncthreads();

  // ---- horizontal 7-tap pass: 22 rows x 16 cols for mu and mu2 ------------
  {
    const float gk[7] = {g0, g1, g2, g3, g2, g1, g0};
#pragma unroll
    for (int it = 0; it < 11; ++it) {
      int idx = lane + 32 * it;               // 0..351
      int row = idx >> 4, col = idx & 15;
      const float* p = &imgT[row * 24 + col];
      float s = 0.0f, s2 = 0.0f;
#pragma unroll
      for (int k = 0; k < 7; ++k) {
        float v = p[k];
        s  += gk[k] * v;
        s2 += gk[k] * (v * v);
      }
      Hmu[row * 16 + col]  = s;
      Hmu2[row * 16 + col] = s2;
    }
    Hmu[352 + lane]  = 0.0f;                  // zero K-pad rows 22,23
    Hmu2[352 + lane] = 0.0f;
  }
  __syncthreads();

  // ---- vertical pass as WMMA: D(16x16) = A(16x24 banded gauss) x H(24x16) -
  // A 16x4 f32 layout: VGPR0 = {K0 | K2}, VGPR1 = {K1 | K3} split at lane 16.
  const int rA   = lane & 15;
  const int koff = (lane < 16) ? 0 : 2;
  const int colB = lane & 15;
  v8f accMu  = {};
  v8f accMu2 = {};
#pragma unroll
  for (int t = 0; t < 6; ++t) {
    int k0 = 4 * t;
    v2f a, bm, bm2;
    a.x = gsel(k0 + koff     - rA, g0, g1, g2, g3);
    a.y = gsel(k0 + koff + 1 - rA, g0, g1, g2, g3);
    bm.x  = Hmu [(k0 + koff) * 16 + colB];
    bm.y  = Hmu [(k0 + koff + 1) * 16 + colB];
    bm2.x = Hmu2[(k0 + koff) * 16 + colB];
    bm2.y = Hmu2[(k0 + koff + 1) * 16 + colB];
    accMu  = __builtin_amdgcn_wmma_f32_16x16x4_f32(false, a, false, bm,
                                                   (short)0, accMu,  false, false);
    accMu2 = __builtin_amdgcn_wmma_f32_16x16x4_f32(false, a, false, bm2,
                                                   (short)0, accMu2, false, false);
  }

  // ---- n = (img - mu) / (sqrt(|mu2 - mu^2|) + 1)  (C/D layout -> LDS) -----
  const int rowOff = (lane < 16) ? 0 : 8;
#pragma unroll
  for (int v = 0; v < 8; ++v) {
    int row = v + rowOff;
    float m  = accMu[v];
    float m2 = accMu2[v];
    float sd = sqrtf(fabsf(m2 - m * m));
    float iv = imgT[(row + 3) * 24 + (colB + 3)];
    nbuf[row * 17 + colB] = (iv - m) / (sd + 1.0f);
  }
  __syncthreads();

  // ---- block variance (ddof=1) over 256 values ----------------------------
  float bs = 0.0f, bs2 = 0.0f;
#pragma unroll
  for (int j = 0; j < 8; ++j) {
    int idx = lane * 8 + j;
    float v = nbuf[(idx >> 4) * 17 + (idx & 15)];
    bs += v; bs2 += v * v;
  }
  bs = wave_sum32(bs); bs2 = wave_sum32(bs2);
  float blockVar = fmaxf(0.0f, bs2 - bs * bs * (1.0f / 256.0f)) * (1.0f / 255.0f);

  // ---- edge impairment: 4 edges x 11 windows of 6, std(ddof=1) < 0.1 ------
  int impFlag = 0;
  for (int t = lane; t < 44; t += 32) {
    int e = t / 11, i = t - e * 11;
    float s = 0.0f, s2 = 0.0f;
#pragma unroll
    for (int j = 0; j < 6; ++j) {
      int idx = i + j;
      float v;
      if (e == 0)      v = nbuf[idx];                 // top row
      else if (e == 1) v = nbuf[idx * 17 + 15];       // right col
      else if (e == 2) v = nbuf[15 * 17 + idx];       // bottom row
      else             v = nbuf[idx * 17];            // left col
      s += v; s2 += v * v;
    }
    float var6 = (s2 - s * s * (1.0f / 6.0f)) * (1.0f / 5.0f);
    if (var6 < 0.01f) impFlag = 1;                    // std < 0.1
  }
  const int impaired = (__ballot(impFlag) != 0ull) ? 1 : 0;

  // ---- center std (cols 7,8 : 32 vals, ddof=1) ----------------------------
  float cv = nbuf[(lane & 15) * 17 + 7 + (lane >> 4)];
  float cs = wave_sum32(cv), cs2 = wave_sum32(cv * cv);
  float stdc = sqrtf(fmaxf(0.0f, cs2 - cs * cs * (1.0f / 32.0f)) * (1.0f / 31.0f));

  // ---- surround std (14 cols {0..6,8,10..15} x 16 rows = 224, ddof=1) -----
  float ss = 0.0f, ss2 = 0.0f;
#pragma unroll
  for (int j = 0; j < 7; ++j) {
    int idx = lane * 7 + j;
    int ci = idx >> 4;
    int col = (ci < 8) ? (int)((0x86543210u >> (4 * ci)) & 15u)
                       : (int)((0x00FEDCBAu >> (4 * (ci - 8))) & 15u);
    float v = nbuf[(idx & 15) * 17 + col];
    ss += v; ss2 += v * v;
  }
  ss = wave_sum32(ss); ss2 = wave_sum32(ss2);
  float stds = sqrtf(fmaxf(0.0f, ss2 - ss * ss * (1.0f / 224.0f)) * (1.0f / 223.0f));

  if (lane == 0) {
    float csd = stdc / stds;
    if (csd != csd) csd = 0.0f;                       // NaN -> 0 (0/0 case)
    float sigma = sqrtf(blockVar);
    float beta  = fabsf(sigma - csd) / fmaxf(sigma, csd);  // may be NaN
    bool active = blockVar > 0.1f;
    bool wnc    = sigma > 2.0f * beta;                // NaN compares false
    float contrib = active ? ((impaired ? (1.0f - blockVar) : 0.0f) +
                              (wnc ? blockVar : 0.0f))
                           : 0.0f;
    int gBlk = (bz * NB + by) * NB + bx;
    contribBuf[gBlk] = contrib;
    activeBuf[gBlk]  = active ? 1.0f : 0.0f;
  }
}

__global__ void piqe_score_kernel(const float* __restrict__ contribBuf,
                                  const float* __restrict__ activeBuf,
                                  float* __restrict__ out) {
  const int b = blockIdx.x, t = threadIdx.x;
  __shared__ float sc[256];
  __shared__ float sa[256];
  float c = 0.0f, a = 0.0f;
  for (int i = t; i < NBLK; i += 256) {              // fixed order: deterministic
    c += contribBuf[b * NBLK + i];
    a += activeBuf[b * NBLK + i];
  }
  sc[t] = c; sa[t] = a;
  __syncthreads();
  for (int s = 128; s > 0; s >>= 1) {
    if (t < s) { sc[t] += sc[t + s]; sa[t] += sa[t + s]; }
    __syncthreads();
  }
  if (t == 0) out[b] = (sc[0] + 1.0f) / (1.0f + sa[0]) * 100.0f;
}

// ---------------------------------------------------------------- launcher
extern "C" void kernel_launch(void* const* d_in, const int* in_sizes, int n_in,
                              void* d_out, int out_size, void* d_ws, size_t ws_size,
                              hipStream_t stream) {
  (void)in_sizes; (void)n_in; (void)out_size; (void)ws_size;
  const float* x = (const float*)d_in[0];
  float* out = (float*)d_out;

  unsigned* maxBits   = (unsigned*)d_ws;                       // 8 u32
  float*    contrib   = (float*)((char*)d_ws + 256);           // 8*16384 f32
  float*    activeArr = contrib + NBATCH * NBLK;               // 8*16384 f32

  piqe_init_kernel<<<1, 32, 0, stream>>>(maxBits);
  piqe_max_kernel<<<dim3(64, NBATCH), 256, 0, stream>>>(x, maxBits);
  piqe_block_kernel<<<dim3(NB, NB, NBATCH), 32, 0, stream>>>(x, maxBits,
                                                             contrib, activeArr);
  piqe_score_kernel<<<NBATCH, 256, 0, stream>>>(contrib, activeArr, out);
}